// DiffRasterizer_83339545412110
// MI455X (gfx1250) — compile-verified
//
#include <hip/hip_runtime.h>
#include <math.h>

typedef float v2f __attribute__((ext_vector_type(2)));
typedef float v8f __attribute__((ext_vector_type(8)));

#define NSHAPES 8
#define NSAMP   30
#define SOFTNESS 0.01f
#define EPS      1e-8f

// workspace float layout
#define WS_POLY  0      // 30 rows x 16 cols (col = m*2 + d), 480 floats
#define WS_ALPHA 480    // 8 sorted effective alphas
#define WS_CSG   488    // 8 sorted csg flags (0/1)
#define WS_COL   496    // 8 x 3 sorted colors
#define WS_TOTAL 520

// basis_full[s][j]: coefficient of control point j for sample s.
// t_global = 4*s/30; seg = clip(floor(t_global),0,3); t = t_global - seg;
// basis = [ti^3, 3ti^2 t, 3ti t^2, t^3] at idx = [3seg,3seg+1,3seg+2,(3seg+3)%12]
__device__ __forceinline__ float basis_at(int s, int j) {
    if (s >= NSAMP) return 0.0f;
    float tg = 4.0f * (float)s / 30.0f;
    int seg = (int)floorf(tg);
    seg = seg < 0 ? 0 : (seg > 3 ? 3 : seg);
    float t  = tg - (float)seg;
    float ti = 1.0f - t;
    float b0 = ti * ti * ti;
    float b1 = 3.0f * ti * ti * t;
    float b2 = 3.0f * ti * t * t;
    float b3 = t * t * t;
    int i0 = 3 * seg, i1 = 3 * seg + 1, i2 = 3 * seg + 2, i3 = (3 * seg + 3) % 12;
    float v = 0.0f;
    if (j == i0) v += b0;
    if (j == i1) v += b1;
    if (j == i2) v += b2;
    if (j == i3) v += b3;
    return v;
}

// One wave32. Sorts shapes by z, evaluates the bezier->polyline einsum as
// poly = B(30x12) @ cp(12x16packed) via chained V_WMMA_F32_16X16X4_F32,
// and stages sorted per-shape params into the workspace.
__global__ __launch_bounds__(32)
void DiffRaster_prep_kernel(const float* __restrict__ P,
                            const float* __restrict__ c,
                            const float* __restrict__ alpha,
                            const float* __restrict__ alive,
                            const float* __restrict__ z,
                            const int*   __restrict__ csg,
                            float* __restrict__ ws)
{
    const int lane = threadIdx.x;          // 0..31, full wave
    // ---- stable argsort of z (8 elements, redundant per lane) ----
    float zl[NSHAPES];
#pragma unroll
    for (int i = 0; i < NSHAPES; ++i) zl[i] = z[i];
    int order[NSHAPES];
#pragma unroll
    for (int i = 0; i < NSHAPES; ++i) {
        int r = 0;
#pragma unroll
        for (int j = 0; j < NSHAPES; ++j)
            if (zl[j] < zl[i] || (zl[j] == zl[i] && j < i)) ++r;
        order[r] = i;
    }

    // ---- WMMA fragments (ISA 7.12.2 layouts, wave32) ----
    const int half = lane >> 4;     // 0: lanes 0-15, 1: lanes 16-31
    const int nidx = lane & 15;     // A: row-in-tile, B/D: column N
    const int mB   = nidx >> 1;     // shape (sorted position) for column N
    const int dB   = nidx & 1;      // 0=x, 1=y
    const int srcB = order[mB];     // unsorted shape index feeding column N

#pragma unroll
    for (int tm = 0; tm < 2; ++tm) {            // two 16-row tiles of M=30
        v8f acc = {};
#pragma unroll
        for (int kc = 0; kc < 3; ++kc) {        // K = 12 as 3 chunks of 4
            const int k0 = kc * 4;
            // A 16x4 f32: v0 = K (k0 | k0+2), v1 = K (k0+1 | k0+3) per lane-half
            v2f a;
            const int M = tm * 16 + nidx;
            a.x = basis_at(M, k0 + 2 * half);
            a.y = basis_at(M, k0 + 2 * half + 1);
            // B 4x16 f32, mirrored K-split across lane halves; N striped on lanes
            v2f b;
            const int kB = k0 + 2 * half;
            b.x = P[srcB * 24 + kB * 2 + dB];
            b.y = P[srcB * 24 + (kB + 1) * 2 + dB];
            acc = __builtin_amdgcn_wmma_f32_16x16x4_f32(
                      false, a, false, b, (short)0, acc, false, false);
        }
        // D 16x16 f32: VGPR v holds row (v | v+8) per lane-half, col = nidx
#pragma unroll
        for (int v = 0; v < 8; ++v) {
            const int s = tm * 16 + v + 8 * half;
            if (s < NSAMP) ws[WS_POLY + s * 16 + nidx] = acc[v];
        }
    }

    // ---- sorted per-shape params ----
    if (lane < NSHAPES) {
        const int src = order[lane];
        const float sg = 1.0f / (1.0f + expf(-alive[src]));
        ws[WS_ALPHA + lane] = (sg > 0.1f) ? alpha[src] : 0.0f;
        ws[WS_CSG   + lane] = csg[src] ? 1.0f : 0.0f;
        ws[WS_COL + lane * 3 + 0] = c[src * 3 + 0];
        ws[WS_COL + lane * 3 + 1] = c[src * 3 + 1];
        ws[WS_COL + lane * 3 + 2] = c[src * 3 + 2];
    }
}

// One thread per pixel. Shape table staged in LDS per workgroup.
__global__ __launch_bounds__(256)
void DiffRaster_raster_kernel(const float* __restrict__ ws,
                              const int*   __restrict__ pwidth,
                              const int*   __restrict__ pheight,
                              float* __restrict__ out,
                              int npix)
{
    __shared__ float sm[WS_TOTAL];
    for (int i = threadIdx.x; i < WS_TOTAL; i += blockDim.x) sm[i] = ws[i];
    __syncthreads();

    const int p = blockIdx.x * blockDim.x + threadIdx.x;
    const int W = pwidth[0];
    const int H = pheight[0];
    if (p >= npix) return;

    const int h = p / W;
    const int w = p - h * W;
    const float px = (W > 1) ? (float)w / (float)(W - 1) : 0.0f;
    const float py = (H > 1) ? (float)h / (float)(H - 1) : 0.0f;

    float rgb0 = 0.0f, rgb1 = 0.0f, rgb2 = 0.0f, ca = 0.0f;

#pragma unroll 1
    for (int m = 0; m < NSHAPES; ++m) {
        float mind2 = 3.4e38f;
        float wn = 0.0f;
#pragma unroll 1
        for (int i = 0; i < NSAMP; ++i) {
            const int i2 = (i + 1 == NSAMP) ? 0 : i + 1;
            const float ax = sm[WS_POLY + i  * 16 + 2 * m];
            const float ay = sm[WS_POLY + i  * 16 + 2 * m + 1];
            const float bx = sm[WS_POLY + i2 * 16 + 2 * m];
            const float by = sm[WS_POLY + i2 * 16 + 2 * m + 1];
            const float abx = bx - ax, aby = by - ay;
            const float apx = px - ax, apy = py - ay;
            float t = (apx * abx + apy * aby) / (abx * abx + aby * aby + EPS);
            t = fminf(fmaxf(t, 0.0f), 1.0f);
            const float dx = apx - t * abx;
            const float dy = apy - t * aby;
            const float d2 = dx * dx + dy * dy;
            mind2 = fminf(mind2, d2);
            const bool up   = (ay <= py) && (py < by);
            const bool dn   = (ay >  py) && (py >= by);
            const bool left = (abx * (py - ay) - apx * (by - ay)) > 0.0f;
            if (up && left)  wn += 1.0f;
            if (dn && !left) wn -= 1.0f;
        }
        const float mind = sqrtf(mind2 + EPS);
        const float sdf  = (wn != 0.0f) ? -mind : mind;
        const float cov  = 1.0f / (1.0f + expf(sdf / SOFTNESS));
        const float la   = cov * sm[WS_ALPHA + m];

        // compositing step (matches the scan in the reference)
        const bool  is_csg = sm[WS_CSG + m] != 0.0f;
        const float one_m  = 1.0f - la;
        const float csg_a  = ca * one_m;
        const float gate   = (csg_a > 0.0f) ? 1.0f : 0.0f;
        const float out_a  = la + ca * one_m;
        const float safe   = (out_a > 0.0f) ? out_a : 1.0f;
        const float c0 = sm[WS_COL + m * 3 + 0];
        const float c1 = sm[WS_COL + m * 3 + 1];
        const float c2 = sm[WS_COL + m * 3 + 2];
        float o0 = (c0 * la + rgb0 * ca * one_m) / safe;
        float o1 = (c1 * la + rgb1 * ca * one_m) / safe;
        float o2 = (c2 * la + rgb2 * ca * one_m) / safe;
        o0 = (out_a > 0.0f) ? o0 : 0.0f;
        o1 = (out_a > 0.0f) ? o1 : 0.0f;
        o2 = (out_a > 0.0f) ? o2 : 0.0f;
        rgb0 = is_csg ? rgb0 * gate : o0;
        rgb1 = is_csg ? rgb1 * gate : o1;
        rgb2 = is_csg ? rgb2 * gate : o2;
        ca   = is_csg ? csg_a : out_a;
    }

    out[p * 3 + 0] = fminf(fmaxf(rgb0 * ca, 0.0f), 1.0f);
    out[p * 3 + 1] = fminf(fmaxf(rgb1 * ca, 0.0f), 1.0f);
    out[p * 3 + 2] = fminf(fmaxf(rgb2 * ca, 0.0f), 1.0f);
}

extern "C" void kernel_launch(void* const* d_in, const int* in_sizes, int n_in,
                              void* d_out, int out_size, void* d_ws, size_t ws_size,
                              hipStream_t stream)
{
    (void)in_sizes; (void)n_in; (void)ws_size;
    const float* P     = (const float*)d_in[0];   // (8,12,2)
    const float* c     = (const float*)d_in[1];   // (8,3)
    const float* alpha = (const float*)d_in[2];   // (8,)
    const float* alive = (const float*)d_in[3];   // (8,)
    const float* z     = (const float*)d_in[4];   // (8,)
    const int*   csg   = (const int*)d_in[5];     // (8,) bool (zeros)
    const int*   pw    = (const int*)d_in[6];     // width scalar
    const int*   ph    = (const int*)d_in[7];     // height scalar

    float* ws  = (float*)d_ws;
    float* out = (float*)d_out;
    const int npix = out_size / 3;

    hipLaunchKernelGGL(DiffRaster_prep_kernel, dim3(1), dim3(32), 0, stream,
                       P, c, alpha, alive, z, csg, ws);
    const int blocks = (npix + 255) / 256;
    hipLaunchKernelGGL(DiffRaster_raster_kernel, dim3(blocks), dim3(256), 0, stream,
                       ws, pw, ph, out, npix);
}